// TextEmbedding_16681652978415
// MI455X (gfx1250) — compile-verified
//
#include <hip/hip_runtime.h>
#include <cstdint>

#define BB  16
#define NTT 4096
#define DD  512

// Kernel 1: per-batch valid-token count (L). One block per batch row.
__global__ __launch_bounds__(256) void te_len_kernel(const long long* __restrict__ text,
                                                     int* __restrict__ lens) {
  __shared__ int cnt[256];
  const int b = blockIdx.x;
  const int t = threadIdx.x;
  const long long* row = text + (size_t)b * NTT;
  int c = 0;
  for (int i = t; i < NTT; i += 256) c += (row[i] >= 0) ? 1 : 0;
  cnt[t] = c;
  __syncthreads();
  for (int s = 128; s > 0; s >>= 1) {
    if (t < s) cnt[t] += cnt[t + s];
    __syncthreads();
  }
  if (t == 0) lens[b] = cnt[0];
}

// Kernel 2: one wave32 per output row. Stretch-gather embedding rows via the
// CDNA5 async DMA path: global -> LDS -> global, tracked on ASYNCcnt.
__global__ __launch_bounds__(256) void te_gather_kernel(const long long* __restrict__ text,
                                                        const float* __restrict__ emb,
                                                        const int* __restrict__ lens,
                                                        float* __restrict__ out) {
  extern __shared__ float smem[];              // 8 waves * 512 floats = 16 KB
  const int wave = threadIdx.x >> 5;
  const int lane = threadIdx.x & 31;
  const int blocksPerBatch = NTT / 8;          // 8 output rows per block
  const int b = blockIdx.x / blocksPerBatch;
  const unsigned p = (unsigned)(blockIdx.x % blocksPerBatch) * 8u + (unsigned)wave;

  float* dst = out + ((size_t)b * NTT + p) * DD;
  const unsigned L = (unsigned)lens[b];

  if (L == 0u) {                               // reference: zero output row
#pragma unroll
    for (int k = 0; k < 4; ++k) {
      float4 z = make_float4(0.f, 0.f, 0.f, 0.f);
      *reinterpret_cast<float4*>(dst + k * 128 + lane * 4) = z;
    }
    return;
  }

  // j(p): each of the first L tokens repeated base or base+1 times.
  const unsigned base = (unsigned)NTT / L;
  const unsigned rem  = (unsigned)NTT % L;
  const unsigned nlo  = L - rem;
  const unsigned boundary = nlo * base;
  const unsigned j = (p < boundary) ? (p / base)
                                    : (nlo + (p - boundary) / (base + 1u));

  const long long tok = text[(size_t)b * NTT + j];   // tail padding => j-th valid == text[b][j]
  const unsigned rowIdx = (unsigned)(tok + 1);        // emb row (0 = pad row, unreachable here)
  const float* src = emb + (size_t)rowIdx * DD;

  // Low 32 bits of the generic pointer to __shared__ == LDS byte offset.
  const unsigned ldsBase = (unsigned)(uintptr_t)(&smem[wave * DD]);

  // 2 KB row: 4 chunks x (32 lanes x 16B) coalesced, async global->LDS.
#pragma unroll
  for (int k = 0; k < 4; ++k) {
    const unsigned ldsAddr = ldsBase + (unsigned)(k * 512 + lane * 16);
    const unsigned long long g =
        (unsigned long long)(uintptr_t)(src + k * 128 + lane * 4);
    asm volatile("global_load_async_to_lds_b128 %0, %1, off"
                 :: "v"(ldsAddr), "v"(g)
                 : "memory");
  }
  asm volatile("s_wait_asynccnt 0x0" ::: "memory");   // LDS writes visible
#pragma unroll
  for (int k = 0; k < 4; ++k) {
    const unsigned ldsAddr = ldsBase + (unsigned)(k * 512 + lane * 16);
    const unsigned long long g =
        (unsigned long long)(uintptr_t)(dst + k * 128 + lane * 4);
    asm volatile("global_store_async_from_lds_b128 %0, %1, off"
                 :: "v"(g), "v"(ldsAddr)
                 : "memory");
  }
  // s_endpgm performs an implicit wait-idle, draining outstanding async stores.
}

extern "C" void kernel_launch(void* const* d_in, const int* in_sizes, int n_in,
                              void* d_out, int out_size, void* d_ws, size_t ws_size,
                              hipStream_t stream) {
  (void)in_sizes; (void)n_in; (void)out_size; (void)ws_size;
  const long long* text = (const long long*)d_in[0];   // int64 (B, NT)
  // d_in[1] = seq_len (== NTT, compile-time constant here)
  const float* emb = (const float*)d_in[2];            // fp32 (V+1, D)
  float* out = (float*)d_out;                          // fp32 (B, NT, D)
  int* lens = (int*)d_ws;                              // B ints of scratch

  te_len_kernel<<<BB, 256, 0, stream>>>(text, lens);
  te_gather_kernel<<<BB * (NTT / 8), 256, 8 * DD * sizeof(float), stream>>>(
      text, emb, lens, out);
}